// Loss_Antonymy_44487271252809
// MI455X (gfx1250) — compile-verified
//
#include <hip/hip_runtime.h>
#include <hip/hip_bf16.h>
#include <stdint.h>

typedef __attribute__((ext_vector_type(2))) float v2f;
typedef __attribute__((ext_vector_type(8))) float v8f;
typedef int v4i __attribute__((vector_size(16)));   // matches builtin's pointee type

#define AS1 __attribute__((address_space(1)))
#define AS3 __attribute__((address_space(3)))

#define THRESH    0.8f
#define ROWPAD    132          // 128 + 4 float pad -> conflict-free 16x16x4 A-layout reads
#define NB        2048         // one wave per block

// ---- CDNA5 async global->LDS copy (ASYNCcnt path) -------------------------
__device__ __forceinline__ void async_b128(const float* g, float* l) {
#if __has_builtin(__builtin_amdgcn_global_load_async_to_lds_b128)
    __builtin_amdgcn_global_load_async_to_lds_b128(
        (AS1 v4i*)g, (AS3 v4i*)l, 0, 0);
#else
    asm volatile("global_load_async_to_lds_b128 %0, %1, off"
                 :: "v"((uint32_t)(uintptr_t)l),
                    "v"((uint64_t)(uintptr_t)g)
                 : "memory");
#endif
}

template <int N>
__device__ __forceinline__ void wait_async() {
#if __has_builtin(__builtin_amdgcn_s_wait_asynccnt)
    __builtin_amdgcn_s_wait_asynccnt(N);
#else
    asm volatile("s_wait_asynccnt %0" :: "i"(N) : "memory");
#endif
}

// ---- main streaming kernel: 1 wave / block, 16-row tiles, double-buffered --
__global__ __launch_bounds__(32) void antonymy_main(
    const float* __restrict__ S2, const float* __restrict__ A1,
    const float* __restrict__ score, float* __restrict__ partial, int n)
{
    __shared__ float lds[2][2][16 * ROWPAD];   // [buf][matrix][row*pitch] = 33,792 B
    const int lane  = threadIdx.x;
    const int tiles = n >> 4;
    const int l4    = lane << 2;               // this lane's 16B slice of a 512B row
    float acc = 0.f;

    auto stage = [&](int t, int b) {
        const float* gS2 = S2 + (size_t)t * 2048;   // 16 rows * 128 cols
        const float* gA1 = A1 + (size_t)t * 2048;
#pragma unroll
        for (int r = 0; r < 16; ++r) {              // 1 async instr = 1 full row (coalesced 512B)
            async_b128(gS2 + r * 128 + l4, &lds[b][0][r * ROWPAD + l4]);
            async_b128(gA1 + r * 128 + l4, &lds[b][1][r * ROWPAD + l4]);
        }
    };

    int buf = 0;
    if (blockIdx.x < tiles) stage(blockIdx.x, 0);

    for (int t = blockIdx.x; t < tiles; t += NB) {
        const int tn = t + NB;
        if (tn < tiles) { stage(tn, buf ^ 1); wait_async<32>(); }  // current tile done, next in flight
        else            { wait_async<0>(); }

        // A-matrix 16x4 f32 layout: lane = h*16 + m holds row m, K = 2h, 2h+1
        const int m = lane & 15, h = lane >> 4;
        const float* s2r = &lds[buf][0][m * ROWPAD + 2 * h];
        const float* a1r = &lds[buf][1][m * ROWPAD + 2 * h];

        v8f c = {0.f, 0.f, 0.f, 0.f, 0.f, 0.f, 0.f, 0.f};
        v2f ones; ones[0] = 1.f; ones[1] = 1.f;     // B = ones => D(m,n) = rowsum(A)
#pragma unroll
        for (int k = 0; k < 32; ++k) {              // 32 K-chunks of 4 cover D=128
            const float2 x = *(const float2*)(s2r + 4 * k);
            const float2 y = *(const float2*)(a1r + 4 * k);
            v2f a;
            a[0] = (y.x - x.x) * (y.x - x.x);
            a[1] = (y.y - x.y) * (y.y - x.y);
            c = __builtin_amdgcn_wmma_f32_16x16x4_f32(
                    false, a, false, ones, (short)0, c, false, false);
        }

        // C layout: VGPR r, lanes 0-15 -> row r; lanes 16-31 -> row 8+r (replicated over N).
        if (m == 0) {                               // lanes 0 and 16 each own 8 rows
            const float* sc = score + (size_t)t * 16 + h * 8;
#pragma unroll
            for (int r = 0; r < 8; ++r) {
                float d  = sqrtf(c[r]);
                float th = tanhf(d);
                acc += (sc[r] >= THRESH) ? fmaxf(1.f - th, 0.f)
                                         : fmaxf(1.f + th, 0.f);
            }
        }
        buf ^= 1;
    }

    float other = __shfl(acc, lane ^ 16, 32);       // combine lane0 + lane16
    if (lane == 0) partial[blockIdx.x] = acc + other;
}

// ---- deterministic finalize: fixed-order strided + tree reduction ----------
__global__ __launch_bounds__(256) void antonymy_reduce(
    const float* __restrict__ partial, int nb,
    const float* __restrict__ S2, const float* __restrict__ A1,
    const float* __restrict__ score, int n, float* __restrict__ out)
{
    __shared__ float sh[256];
    const int tid = threadIdx.x;
    float s = 0.f;
    for (int i = tid; i < nb; i += 256) s += partial[i];
    sh[tid] = s;
    __syncthreads();
    for (int w = 128; w > 0; w >>= 1) {
        if (tid < w) sh[tid] += sh[tid + w];
        __syncthreads();
    }
    if (tid == 0) {
        float total = sh[0];
        for (int r = (n & ~15); r < n; ++r) {       // tail rows (empty when 16 | n)
            float ss = 0.f;
            for (int k = 0; k < 128; ++k) {
                float d = A1[(size_t)r * 128 + k] - S2[(size_t)r * 128 + k];
                ss += d * d;
            }
            float th = tanhf(sqrtf(ss));
            total += (score[r] >= THRESH) ? fmaxf(1.f - th, 0.f)
                                          : fmaxf(1.f + th, 0.f);
        }
        out[0] = total / (float)n;
    }
}

extern "C" void kernel_launch(void* const* d_in, const int* in_sizes, int n_in,
                              void* d_out, int out_size, void* d_ws, size_t ws_size,
                              hipStream_t stream) {
    const float* S2 = (const float*)d_in[0];
    const float* A1 = (const float*)d_in[1];
    const float* sc = (const float*)d_in[2];
    float* out      = (float*)d_out;
    float* partial  = (float*)d_ws;                 // NB floats = 8 KB scratch
    const int n     = in_sizes[2];                  // rows = score count

    antonymy_main<<<NB, 32, 0, stream>>>(S2, A1, sc, partial, n);
    antonymy_reduce<<<1, 256, 0, stream>>>(partial, NB, S2, A1, sc, n, out);
}